// DIoUAnswerSpanLoss_9423158247590
// MI455X (gfx1250) — compile-verified
//
#include <hip/hip_runtime.h>
#include <stdint.h>

// Problem constants (match reference: B=32, L=2048)
#define TPB   512
#define LSEQ  2048
#define IPT   (LSEQ / TPB)      // 4 contiguous elements per thread
#define NEGV  (-10000.0f)
#define NINF  (-__builtin_inff())
#define IMAX  0x7fffffff

// ---- CDNA5 async global->LDS staging (ASYNCcnt path), guarded ------------
#if defined(__HIP_DEVICE_COMPILE__) && defined(__gfx1250__)
#if __has_builtin(__builtin_amdgcn_global_load_async_to_lds_b128)
#define USE_ASYNC_LDS 1
#endif
#endif

#ifdef USE_ASYNC_LDS
// Builtin signature (from hipcc diagnostic): pointee is int4 vector,
// src in addrspace(1) (global), dst in addrspace(3) (LDS).
typedef int v4i __attribute__((vector_size(16)));
typedef __attribute__((address_space(1))) v4i* gv4i_p;   // pointer to AS1 v4i
typedef __attribute__((address_space(3))) v4i* lv4i_p;   // pointer to AS3 v4i
#endif

__device__ __forceinline__ void stage16(const float* g, float* l) {
#ifdef USE_ASYNC_LDS
  // Per-lane 16B DMA into LDS; low 32 bits of a generic LDS pointer are the
  // workgroup-relative LDS byte address on AMDGPU.
  __builtin_amdgcn_global_load_async_to_lds_b128(
      (gv4i_p)(unsigned long long)(uintptr_t)g,
      (lv4i_p)(unsigned)(uintptr_t)l,
      0, 0);
#else
  *(float4*)l = *(const float4*)g;
#endif
}

__device__ __forceinline__ void wait_async_lds() {
#ifdef USE_ASYNC_LDS
#if __has_builtin(__builtin_amdgcn_s_wait_asynccnt)
  __builtin_amdgcn_s_wait_asynccnt(0);
#else
  asm volatile("s_wait_asynccnt 0" ::: "memory");
#endif
#endif
}

// ---- Kernel 1: one workgroup per batch row -------------------------------
// Computes sp[b], ep[b] = argmax over valid (start<=end) pairs of
// log_softmax(start)[i] + log_softmax(end)[j], with first-occurrence
// argmax tie-breaking (lexicographic (max value, min index) monoid).
extern "C" __global__ void __launch_bounds__(TPB)
diou_row_kernel(const int* __restrict__ c_ids,
                const float* __restrict__ slog,
                const float* __restrict__ elog,
                int2* __restrict__ spep)
{
  __shared__ __align__(16) float s_a[LSEQ];   // start row, reused for prefix-max values
  __shared__ __align__(16) float s_b[LSEQ];   // end row
  __shared__ int   s_pmi[LSEQ];               // prefix argmax index
  __shared__ float s_rv[TPB];                 // reduce/scan scratch (values)
  __shared__ float s_rw[TPB];                 // reduce scratch (2nd values)
  __shared__ int   s_ri[TPB];                 // reduce/scan scratch (indices)

  const int b = blockIdx.x;
  const int t = threadIdx.x;
  const int base = t * IPT;

  const float* srow = slog + (size_t)b * LSEQ;
  const float* erow = elog + (size_t)b * LSEQ;
  const int*   mrow = c_ids + (size_t)b * LSEQ;

  // Warm the mask row into cache while the async DMAs run.
  __builtin_prefetch(mrow + base, 0, 0);

  // Stage both logit rows into LDS (one b128 per row per lane = exactly 8KB/row).
  stage16(srow + base, &s_a[base]);
  stage16(erow + base, &s_b[base]);
  wait_async_lds();
  __syncthreads();

  // ---- logsumexp of both rows (two fused block reductions) ----
  float mxs = NINF, mxe = NINF;
#pragma unroll
  for (int k = 0; k < IPT; ++k) {
    mxs = fmaxf(mxs, s_a[base + k]);
    mxe = fmaxf(mxe, s_b[base + k]);
  }
  s_rv[t] = mxs; s_rw[t] = mxe;
  __syncthreads();
  for (int off = TPB / 2; off > 0; off >>= 1) {
    if (t < off) {
      s_rv[t] = fmaxf(s_rv[t], s_rv[t + off]);
      s_rw[t] = fmaxf(s_rw[t], s_rw[t + off]);
    }
    __syncthreads();
  }
  const float Ms = s_rv[0], Me = s_rw[0];
  __syncthreads();

  float sus = 0.f, sue = 0.f;
#pragma unroll
  for (int k = 0; k < IPT; ++k) {
    sus += expf(s_a[base + k] - Ms);
    sue += expf(s_b[base + k] - Me);
  }
  s_rv[t] = sus; s_rw[t] = sue;
  __syncthreads();
  for (int off = TPB / 2; off > 0; off >>= 1) {
    if (t < off) { s_rv[t] += s_rv[t + off]; s_rw[t] += s_rw[t + off]; }
    __syncthreads();
  }
  const float lseS = Ms + logf(s_rv[0]);
  const float lseE = Me + logf(s_rw[0]);
  __syncthreads();

  // ---- inclusive prefix (max, first-index) scan of masked start log-probs ----
  float lv[IPT]; int li[IPT];
  float rv = NINF; int ri = IMAX;
#pragma unroll
  for (int k = 0; k < IPT; ++k) {
    const int i = base + k;
    const float x = (mrow[i] != 0) ? (s_a[i] - lseS) : NINF;
    if (x > rv || (x == rv && i < ri)) { rv = x; ri = i; }
    lv[k] = rv; li[k] = ri;
  }
  s_rv[t] = rv; s_ri[t] = ri;
  __syncthreads();
  // Hillis-Steele over per-thread totals
  for (int off = 1; off < TPB; off <<= 1) {
    float pv = NINF; int pi = IMAX;
    const bool has = (t >= off);
    if (has) { pv = s_rv[t - off]; pi = s_ri[t - off]; }
    __syncthreads();
    if (has) {
      const float cv = s_rv[t]; const int ci = s_ri[t];
      if (pv > cv || (pv == cv && pi < ci)) { s_rv[t] = pv; s_ri[t] = pi; }
    }
    __syncthreads();
  }
  float ev = NINF; int ei = IMAX;                 // exclusive prefix for this thread
  if (t > 0) { ev = s_rv[t - 1]; ei = s_ri[t - 1]; }
#pragma unroll
  for (int k = 0; k < IPT; ++k) {
    float pv; int pi;
    if (ev > lv[k] || (ev == lv[k] && ei < li[k])) { pv = ev; pi = ei; }
    else { pv = lv[k]; pi = li[k]; }
    s_a[base + k] = pv;                           // reuse s_a: prefix-max values
    s_pmi[base + k] = pi;
  }
  __syncthreads();

  // ---- argmax over end position j of best_per_end[j] ----
  float bv = NINF; int bj = IMAX;
#pragma unroll
  for (int k = 0; k < IPT; ++k) {
    const int j = base + k;
    const float cand = (mrow[j] != 0) ? (s_a[j] + (s_b[j] - lseE)) : NEGV;
    if (cand > bv || (cand == bv && j < bj)) { bv = cand; bj = j; }
  }
  s_rv[t] = bv; s_ri[t] = bj;
  __syncthreads();
  for (int off = TPB / 2; off > 0; off >>= 1) {
    if (t < off) {
      const float ov = s_rv[t + off]; const int oi = s_ri[t + off];
      if (ov > s_rv[t] || (ov == s_rv[t] && oi < s_ri[t])) { s_rv[t] = ov; s_ri[t] = oi; }
    }
    __syncthreads();
  }
  if (t == 0) {
    const int end = s_ri[0];
    const int sp = (mrow[end] != 0) ? s_pmi[end] : 0;  // all-NEG column -> argmax==0
    spep[b] = make_int2(sp, end);
  }
}

// ---- Kernel 2: single wave32 computes the DIoU scalar --------------------
extern "C" __global__ void __launch_bounds__(32)
diou_finish_kernel(const int2* __restrict__ spep,
                   const int* __restrict__ gsp,
                   const int* __restrict__ gep,
                   float* __restrict__ out, int B)
{
  const int t = threadIdx.x;
  const bool act = (t < B);
  float sp = 0.f, ep = 0.f, g0 = 0.f, g1 = 0.f;
  if (act) {
    const int2 se = spep[t];
    sp = (float)se.x; ep = (float)se.y;
    g0 = (float)gsp[t]; g1 = (float)gep[t];
  }
  const float cd  = (ep - sp + 1.f) * 0.5f;
  const float gcd = (g1 - g0 + 1.f) * 0.5f;
  float num  = act ? (cd - gcd) * (cd - gcd) : 0.f;
  const float dmn = fmaxf(ep, g1) - fminf(sp, g0);
  float den  = act ? dmn * dmn : 0.f;
  float usum = act ? (ep - sp) + (g1 - g0) : 0.f;
  // Reference quirk: intersection comes from batch element 0 only.
  const float inter = __shfl(fminf(ep, g1) - fmaxf(sp, g0), 0);
#pragma unroll
  for (int off = 16; off > 0; off >>= 1) {   // wave32 tree (warpSize == 32)
    num  += __shfl_down(num, off);
    den  += __shfl_down(den, off);
    usum += __shfl_down(usum, off);
  }
  if (t == 0) {
    const float uni = usum - (float)B * inter;
    out[0] = 1.f - inter / uni + num / den;
  }
}

// ---- Host launcher -------------------------------------------------------
extern "C" void kernel_launch(void* const* d_in, const int* in_sizes, int n_in,
                              void* d_out, int out_size, void* d_ws, size_t ws_size,
                              hipStream_t stream) {
  const int*   c_ids = (const int*)d_in[0];
  const int*   gsp   = (const int*)d_in[1];
  const int*   gep   = (const int*)d_in[2];
  const float* slog  = (const float*)d_in[3];
  const float* elog  = (const float*)d_in[4];
  const int B = in_sizes[1];           // gt_start_positions has B elements

  int2* spep = (int2*)d_ws;            // B * 8 bytes of scratch

  diou_row_kernel<<<B, TPB, 0, stream>>>(c_ids, slog, elog, spep);
  diou_finish_kernel<<<1, 32, 0, stream>>>(spep, gsp, gep, (float*)d_out, B);
}